// AutoSparse_36532991820369
// MI455X (gfx1250) — compile-verified
//
#include <hip/hip_runtime.h>
#include <hip/hip_bf16.h>

// AutoSparse forward: out = sign(W) * relu(|W| - sigmoid(threshold[row]))
// W: [4096, 8192] fp32, threshold: [4096, 1] fp32, out: [4096, 8192] fp32.
// Memory-bound streaming kernel: 256 MiB single-touch traffic -> NT hints,
// B128 vector access, one sigmoid per thread per row.

typedef __attribute__((ext_vector_type(4))) float v4f;

__global__ __launch_bounds__(256) void autosparse_fwd_kernel(
    const float* __restrict__ weight,
    const float* __restrict__ threshold,
    float* __restrict__ out,
    int cols) {
  const int row = blockIdx.x;

  // Row-uniform threshold: lowers to scalar load + one v_exp_f32 per thread.
  const float t = threshold[row];
  const float s = 1.0f / (1.0f + expf(-t));   // sigmoid(threshold[row])

  const v4f* __restrict__ wrow =
      reinterpret_cast<const v4f*>(weight + (size_t)row * (size_t)cols);
  v4f* __restrict__ orow =
      reinterpret_cast<v4f*>(out + (size_t)row * (size_t)cols);

  const int nvec = cols >> 2;          // 8192 / 4 = 2048 float4 per row
  const int stride = blockDim.x;       // 256

  for (int i = threadIdx.x; i < nvec; i += stride) {
    // Prefetch next iteration's cacheline (global_prefetch_b8; speculative,
    // LOADcnt-free, safe to run past the row end).
    __builtin_prefetch(&wrow[i + stride], 0, 0);

    // Single-touch streaming data: non-temporal B128 load.
    v4f v = __builtin_nontemporal_load(&wrow[i]);

    v4f r;
    // sign(w)*relu(|w|-s): relu result >= 0, so copysign reproduces
    // sign()-multiply semantics (w==0 -> |w|-s < 0 -> relu==0 -> out 0).
    r.x = __builtin_copysignf(fmaxf(__builtin_fabsf(v.x) - s, 0.0f), v.x);
    r.y = __builtin_copysignf(fmaxf(__builtin_fabsf(v.y) - s, 0.0f), v.y);
    r.z = __builtin_copysignf(fmaxf(__builtin_fabsf(v.z) - s, 0.0f), v.z);
    r.w = __builtin_copysignf(fmaxf(__builtin_fabsf(v.w) - s, 0.0f), v.w);

    // Output is never re-read on device: non-temporal B128 store.
    __builtin_nontemporal_store(r, &orow[i]);
  }
}

extern "C" void kernel_launch(void* const* d_in, const int* in_sizes, int n_in,
                              void* d_out, int out_size, void* d_ws, size_t ws_size,
                              hipStream_t stream) {
  const float* weight    = (const float*)d_in[0];   // [4096*8192]
  const float* threshold = (const float*)d_in[1];   // [4096]
  float* out             = (float*)d_out;           // [4096*8192]

  const int rows = in_sizes[1];                     // 4096 (threshold count)
  const int cols = in_sizes[0] / rows;              // 8192

  // One block per row: 256 threads = 8 wave32s; each thread does
  // cols/4/256 = 8 B128 load/compute/store iterations.
  dim3 grid(rows);
  dim3 block(256);
  autosparse_fwd_kernel<<<grid, block, 0, stream>>>(weight, threshold, out, cols);
}